// GNNClassifier_4200478016020
// MI455X (gfx1250) — compile-verified
//
#include <hip/hip_runtime.h>
#include <hip/hip_bf16.h>

typedef __attribute__((ext_vector_type(2))) float v2f;
typedef __attribute__((ext_vector_type(8))) float v8f;

#define D_FEAT   128
#define HIDDEN   64
#define N_CLASSES 10
#define N_GRAPHS 64

// ---------------------------------------------------------------------------
// Utility: zero a float buffer (graph-capture-safe, deterministic)
// ---------------------------------------------------------------------------
__global__ void zero_f32(float* __restrict__ p, long long n) {
    long long i = (long long)blockIdx.x * blockDim.x + threadIdx.x;
    if (i < n) p[i] = 0.0f;
}

// ---------------------------------------------------------------------------
// In-degree over dst (self-loops added analytically as +1 in dinv kernel)
// ---------------------------------------------------------------------------
__global__ void deg_kernel(const int* __restrict__ dst, float* __restrict__ deg, int E) {
    int e = blockIdx.x * blockDim.x + threadIdx.x;
    if (e < E) atomicAdd(&deg[dst[e]], 1.0f);
}

__global__ void dinv_kernel(float* __restrict__ deg, int N) {
    int i = blockIdx.x * blockDim.x + threadIdx.x;
    if (i < N) {
        float d = deg[i] + 1.0f;   // +1: appended self-loop
        deg[i] = rsqrtf(d);        // d >= 1 always
    }
}

// ---------------------------------------------------------------------------
// WMMA f32 GEMM:  C[M x 64] = A[M x K] @ B[K x 64]
// One wave computes one 16x16 tile of C using V_WMMA_F32_16X16X4_F32.
// 4 waves per block cover the 4 N-tiles (HIDDEN = 64). M must be mult of 16.
//
// A 16x4 f32 layout : lane&15 -> M row;  VGPR{0,1} hold K = 2*(lane>>4)+{0,1}
// B 4x16 f32 layout : lane&15 -> N col;  VGPR{0,1} hold K = 2*(lane>>4)+{0,1}
// C 16x16 f32 layout: VGPR r -> M = r + 8*(lane>>4), N = lane&15
// ---------------------------------------------------------------------------
__global__ __launch_bounds__(128) void gemm_wmma_f32(
    const float* __restrict__ A,   // [M x K] row-major
    const float* __restrict__ B,   // [K x 64] row-major
    float* __restrict__ C,         // [M x 64] row-major
    int K)
{
    const int lane  = threadIdx.x & 31;
    const int wave  = threadIdx.x >> 5;        // 0..3 -> N tile
    const int mbase = blockIdx.x * 16;
    const int nbase = wave * 16;
    const int r15   = lane & 15;               // A: row  | B: col | C: col
    const int khalf = (lane >> 4) * 2;         // K sub-offset within group of 4

    v8f acc = {};
    const float* arow = A + (size_t)(mbase + r15) * K;
    for (int k0 = 0; k0 < K; k0 += 4) {
        v2f a, b;
        const float* ap = arow + k0 + khalf;
        a.x = ap[0];
        a.y = ap[1];
        const float* bp = B + (size_t)(k0 + khalf) * HIDDEN + nbase + r15;
        b.x = bp[0];
        b.y = bp[HIDDEN];
        acc = __builtin_amdgcn_wmma_f32_16x16x4_f32(
            /*neg_a=*/false, a, /*neg_b=*/false, b,
            /*c_mod=*/(short)0, acc, /*reuse_a=*/false, /*reuse_b=*/false);
    }

    const int mhi = (lane >> 4) * 8;
    float* cp = C + (size_t)(mbase + mhi) * HIDDEN + nbase + r15;
#pragma unroll
    for (int r = 0; r < 8; ++r)
        cp[(size_t)r * HIDDEN] = acc[r];
}

// ---------------------------------------------------------------------------
// Edge scatter-add:  agg[dst] += h[src] * dinv[src]*dinv[dst]
// One edge per 32 lanes; 2 features per lane (HIDDEN = 64).
// Edges e >= E are the appended self-loops (s = d = e - E).
// ---------------------------------------------------------------------------
__global__ void scatter_kernel(const float* __restrict__ h,
                               const int* __restrict__ src,
                               const int* __restrict__ dst,
                               const float* __restrict__ dinv,
                               float* __restrict__ agg,
                               int E, int N)
{
    long long t  = (long long)blockIdx.x * blockDim.x + threadIdx.x;
    long long nt = (long long)(E + N) * 32;
    if (t >= nt) return;
    int e = (int)(t >> 5);
    int f = ((int)t & 31) * 2;
    int s, d;
    if (e < E) { s = src[e]; d = dst[e]; }
    else       { s = d = e - E; }
    float nrm = dinv[s] * dinv[d];
    const float* hp = h + (size_t)s * HIDDEN + f;
    float* op = agg + (size_t)d * HIDDEN + f;
    atomicAdd(op,     hp[0] * nrm);
    atomicAdd(op + 1, hp[1] * nrm);
}

// ---------------------------------------------------------------------------
// In-place bias + ReLU over [N x 64]
// ---------------------------------------------------------------------------
__global__ void bias_relu_kernel(float* __restrict__ h,
                                 const float* __restrict__ b, long long n) {
    long long i = (long long)blockIdx.x * blockDim.x + threadIdx.x;
    if (i < n) {
        float v = h[i] + b[(int)(i & (HIDDEN - 1))];
        h[i] = v > 0.0f ? v : 0.0f;
    }
}

// ---------------------------------------------------------------------------
// Global mean-pool accumulation: pooled[g] += h[node], cnt[g] += 1
// ---------------------------------------------------------------------------
__global__ void pool_kernel(const float* __restrict__ h,
                            const int* __restrict__ batch,
                            float* __restrict__ pooled,
                            float* __restrict__ cnt, int N)
{
    long long t = (long long)blockIdx.x * blockDim.x + threadIdx.x;
    if (t >= (long long)N * 32) return;
    int node = (int)(t >> 5);
    int f = ((int)t & 31) * 2;
    int g = batch[node];
    const float* hp = h + (size_t)node * HIDDEN + f;
    atomicAdd(&pooled[g * HIDDEN + f],     hp[0]);
    atomicAdd(&pooled[g * HIDDEN + f + 1], hp[1]);
    if (((int)t & 31) == 0) atomicAdd(&cnt[g], 1.0f);
}

// ---------------------------------------------------------------------------
// Head: logits = (pooled/cnt) @ lin_W + lin_b ; log_softmax.  64 threads.
// ---------------------------------------------------------------------------
__global__ void head_kernel(const float* __restrict__ pooled,
                            const float* __restrict__ cnt,
                            const float* __restrict__ W,   // [64 x 10]
                            const float* __restrict__ b,   // [10]
                            float* __restrict__ out)       // [64 x 10]
{
    int g = threadIdx.x;
    if (g >= N_GRAPHS) return;
    float c = cnt[g];
    c = c > 1.0f ? c : 1.0f;
    float inv = 1.0f / c;
    float logits[N_CLASSES];
#pragma unroll
    for (int j = 0; j < N_CLASSES; ++j) {
        float acc = b[j];
        for (int k = 0; k < HIDDEN; ++k)
            acc += (pooled[g * HIDDEN + k] * inv) * W[k * N_CLASSES + j];
        logits[j] = acc;
    }
    float mx = logits[0];
#pragma unroll
    for (int j = 1; j < N_CLASSES; ++j) mx = fmaxf(mx, logits[j]);
    float sum = 0.0f;
#pragma unroll
    for (int j = 0; j < N_CLASSES; ++j) sum += __expf(logits[j] - mx);
    float lse = mx + __logf(sum);
#pragma unroll
    for (int j = 0; j < N_CLASSES; ++j) out[g * N_CLASSES + j] = logits[j] - lse;
}

// ---------------------------------------------------------------------------
// Launch
// ---------------------------------------------------------------------------
extern "C" void kernel_launch(void* const* d_in, const int* in_sizes, int n_in,
                              void* d_out, int out_size, void* d_ws, size_t ws_size,
                              hipStream_t stream) {
    const float* x     = (const float*)d_in[0];   // [N,128]
    const int*   ei    = (const int*)  d_in[1];   // [2,E]
    const int*   batch = (const int*)  d_in[2];   // [N]
    const float* W1    = (const float*)d_in[3];   // [128,64]
    const float* b1    = (const float*)d_in[4];   // [64]
    const float* W2    = (const float*)d_in[5];   // [64,64]
    const float* b2    = (const float*)d_in[6];   // [64]
    const float* linW  = (const float*)d_in[7];   // [64,10]
    const float* linb  = (const float*)d_in[8];   // [10]
    float* out = (float*)d_out;

    const int N = in_sizes[0] / D_FEAT;           // 100000
    const int E = in_sizes[1] / 2;                // 3200000
    const int* src = ei;
    const int* dst = ei + E;

    // Workspace carve-up (floats), 256B-aligned blocks.
    float* ws = (float*)d_ws;
    float* dinv   = ws;                           // [N]
    float* bufA   = dinv + ((N + 63) & ~63);      // [N*64]
    float* bufB   = bufA + (size_t)N * HIDDEN;    // [N*64]
    float* pooled = bufB + (size_t)N * HIDDEN;    // [64*64]
    float* cnt    = pooled + N_GRAPHS * HIDDEN;   // [64]

    const long long NH = (long long)N * HIDDEN;
    const int ZB = 256;

    // --- degrees -> dinv ---
    zero_f32<<<(N + ZB - 1) / ZB, ZB, 0, stream>>>(dinv, N);
    deg_kernel<<<(E + ZB - 1) / ZB, ZB, 0, stream>>>(dst, dinv, E);
    dinv_kernel<<<(N + ZB - 1) / ZB, ZB, 0, stream>>>(dinv, N);

    // --- layer 1: h = x @ W1 (WMMA) ; agg = scatter(h*norm) ; relu(agg+b1) ---
    gemm_wmma_f32<<<N / 16, 128, 0, stream>>>(x, W1, bufA, D_FEAT);
    zero_f32<<<(NH + ZB - 1) / ZB, ZB, 0, stream>>>(bufB, NH);
    {
        long long nt = (long long)(E + N) * 32;
        scatter_kernel<<<(unsigned)((nt + ZB - 1) / ZB), ZB, 0, stream>>>(
            bufA, src, dst, dinv, bufB, E, N);
    }
    bias_relu_kernel<<<(NH + ZB - 1) / ZB, ZB, 0, stream>>>(bufB, b1, NH);

    // --- layer 2: h = relu1 @ W2 (WMMA) ; agg ; relu(agg+b2) ---
    gemm_wmma_f32<<<N / 16, 128, 0, stream>>>(bufB, W2, bufA, HIDDEN);
    zero_f32<<<(NH + ZB - 1) / ZB, ZB, 0, stream>>>(bufB, NH);
    {
        long long nt = (long long)(E + N) * 32;
        scatter_kernel<<<(unsigned)((nt + ZB - 1) / ZB), ZB, 0, stream>>>(
            bufA, src, dst, dinv, bufB, E, N);
    }
    bias_relu_kernel<<<(NH + ZB - 1) / ZB, ZB, 0, stream>>>(bufB, b2, NH);

    // --- mean pool + head ---
    zero_f32<<<1, 256, 0, stream>>>(pooled, N_GRAPHS * HIDDEN + N_GRAPHS);
    {
        long long nt = (long long)N * 32;
        pool_kernel<<<(unsigned)((nt + ZB - 1) / ZB), ZB, 0, stream>>>(
            bufB, batch, pooled, cnt, N);
    }
    head_kernel<<<1, 64, 0, stream>>>(pooled, cnt, linW, linb, out);
}